// CodeBP_50792283242896
// MI455X (gfx1250) — compile-verified
//
#include <hip/hip_runtime.h>
#include <cstddef>

#define KK 4096
#define NN 8192
#define INFCAP 100.0f
#define EPSV 1e-4f

typedef __attribute__((ext_vector_type(2))) float v2f;
typedef __attribute__((ext_vector_type(8))) float v8f;

// ---------------------------------------------------------------------------
// Wave-level column sum of a 16-column strip over rows [k0, k1) using
// V_WMMA_F32_16X16X4_F32 with A == ones(16x4):
//   D[m,n] = sum_k 1 * B[k,n] + C[m,n]   -> per-column sums, identical in all m.
// Lane L (and L+16) ends up holding the sum for column col_base + (L & 15).
// Two accumulators break the WMMA->WMMA RAW dependency chain.
// ---------------------------------------------------------------------------
__device__ __forceinline__ float wave_colsum16(const float* __restrict__ M,
                                               int col_base, int k0, int k1) {
  const int lane = threadIdx.x & 31;
  const int half = lane >> 4;            // 0: lanes 0-15, 1: lanes 16-31
  const int col  = col_base + (lane & 15);

  v8f acc0 = {};
  v8f acc1 = {};
  v2f ones;
  ones[0] = 1.0f;
  ones[1] = 1.0f;

  const float* p = M + (size_t)(k0 + half) * NN + col;
  for (int k = k0; k < k1; k += 8) {
    v2f b0, b1;
    b0[0] = p[0];                        // rows k+0 / k+1 (by half)
    b0[1] = p[(size_t)2 * NN];           // rows k+2 / k+3
    b1[0] = p[(size_t)4 * NN];           // rows k+4 / k+5
    b1[1] = p[(size_t)6 * NN];           // rows k+6 / k+7
    p += (size_t)8 * NN;
    acc0 = __builtin_amdgcn_wmma_f32_16x16x4_f32(false, ones, false, b0,
                                                 (short)0, acc0, false, false);
    acc1 = __builtin_amdgcn_wmma_f32_16x16x4_f32(false, ones, false, b1,
                                                 (short)0, acc1, false, false);
  }
  return acc0[0] + acc1[0];
}

// ---------------------------------------------------------------------------
// Kernel 1: colsum(Hxs) per column + dope/phi -> clipped prior, dope (to ws).
// grid: N/16 blocks, 256 threads (8 waves); wave w reduces rows [w*512,(w+1)*512).
// ---------------------------------------------------------------------------
__global__ __launch_bounds__(256) void k1_prior(const float* __restrict__ Hxs,
                                                const float* __restrict__ ps,
                                                const float* __restrict__ Min,
                                                float* __restrict__ prior_clip,
                                                float* __restrict__ dope_out) {
  __shared__ float partial[8 * 16];
  const int wave     = threadIdx.x >> 5;
  const int lane     = threadIdx.x & 31;
  const int col_base = blockIdx.x * 16;
  const int rows_per_wave = KK / 8;

  float cs = wave_colsum16(Hxs, col_base, wave * rows_per_wave,
                           (wave + 1) * rows_per_wave);
  if (lane < 16) partial[wave * 16 + lane] = cs;
  __syncthreads();

  if (threadIdx.x < 16) {
    float s = 0.0f;
#pragma unroll
    for (int w = 0; w < 8; ++w) s += partial[w * 16 + threadIdx.x];
    const int n = col_base + threadIdx.x;
    // normalization cancels: log((1-p1/(p0+p1))/(p1/(p0+p1))) = log(p0)-log(p1)
    const float dope = logf(ps[2 * n]) - logf(ps[2 * n + 1]);
    const float phi  = logf(Min[2 * n]) - logf(Min[2 * n + 1]);
    const float prior = dope + phi + s;
    prior_clip[n] = fminf(INFCAP, fmaxf(-INFCAP, prior));
    dope_out[n]   = dope;
  }
}

// ---------------------------------------------------------------------------
// Kernel 2: per-row extrinsic update. One block (256 threads) per row.
// Row tanhm cached in LDS (32 KB) so the leave-one-out pass is LDS-only.
// ---------------------------------------------------------------------------
__device__ __forceinline__ float leave_one_out(float t, float total, float sgn) {
  if (t == 0.0f) return 0.0f;                 // zero_mask -> 2*atanh(0) -> 0
  const float y = sgn * total / t;            // prod / tanhm_mod, signed by (1-2x)
  const float v = logf((1.0f + y) / (1.0f - y)); // == 2*atanh(y)
  return fminf(1.0f, fmaxf(-1.0f, v));
}

__global__ __launch_bounds__(256) void k2_rows(const float* __restrict__ Hxs,
                                               const int* __restrict__ H,
                                               const int* __restrict__ x,
                                               const float* __restrict__ prior_clip,
                                               float* __restrict__ Hxs_new) {
  __shared__ float t_lds[NN];
  __shared__ float red[256];
  const int tid = threadIdx.x;
  const int k   = blockIdx.x;

  const float* hxs_row = Hxs + (size_t)k * NN;
  const int*   h_row   = H + (size_t)k * NN;
  float*       out_row = Hxs_new + (size_t)k * NN;

  float prod = 1.0f;
#pragma unroll
  for (int i = 0; i < NN / 1024; ++i) {
    const int base = i * 1024 + tid * 4;
    const float4 hx = *reinterpret_cast<const float4*>(hxs_row + base);
    const int4   hh = *reinterpret_cast<const int4*>(h_row + base);
    const float4 cp = *reinterpret_cast<const float4*>(prior_clip + base);
    // Hsx = clip(prior)*H - Hxs + EPS*H  ==  (H ? cp+EPS : 0) - Hxs
    const float t0 = tanhf(0.5f * ((hh.x ? (cp.x + EPSV) : 0.0f) - hx.x));
    const float t1 = tanhf(0.5f * ((hh.y ? (cp.y + EPSV) : 0.0f) - hx.y));
    const float t2 = tanhf(0.5f * ((hh.z ? (cp.z + EPSV) : 0.0f) - hx.z));
    const float t3 = tanhf(0.5f * ((hh.w ? (cp.w + EPSV) : 0.0f) - hx.w));
    t_lds[base + 0] = t0;
    t_lds[base + 1] = t1;
    t_lds[base + 2] = t2;
    t_lds[base + 3] = t3;
    prod *= (t0 == 0.0f) ? 1.0f : t0;
    prod *= (t1 == 0.0f) ? 1.0f : t1;
    prod *= (t2 == 0.0f) ? 1.0f : t2;
    prod *= (t3 == 0.0f) ? 1.0f : t3;
  }

  red[tid] = prod;
  __syncthreads();
#pragma unroll
  for (int s = 128; s >= 1; s >>= 1) {
    if (tid < s) red[tid] *= red[tid + s];
    __syncthreads();
  }
  const float total = red[0];
  const float sgn   = 1.0f - 2.0f * (float)x[k];

#pragma unroll
  for (int i = 0; i < NN / 1024; ++i) {
    const int base = i * 1024 + tid * 4;
    float4 o;
    o.x = leave_one_out(t_lds[base + 0], total, sgn);
    o.y = leave_one_out(t_lds[base + 1], total, sgn);
    o.z = leave_one_out(t_lds[base + 2], total, sgn);
    o.w = leave_one_out(t_lds[base + 3], total, sgn);
    *reinterpret_cast<float4*>(out_row + base) = o;
  }
}

// ---------------------------------------------------------------------------
// Kernel 3: colsum(Hxs_new) via WMMA -> output beliefs M_out[N,2].
// ---------------------------------------------------------------------------
__global__ __launch_bounds__(256) void k3_beliefs(const float* __restrict__ Hxs_new,
                                                  const float* __restrict__ dope,
                                                  float* __restrict__ M_out) {
  __shared__ float partial[8 * 16];
  const int wave     = threadIdx.x >> 5;
  const int lane     = threadIdx.x & 31;
  const int col_base = blockIdx.x * 16;
  const int rows_per_wave = KK / 8;

  float cs = wave_colsum16(Hxs_new, col_base, wave * rows_per_wave,
                           (wave + 1) * rows_per_wave);
  if (lane < 16) partial[wave * 16 + lane] = cs;
  __syncthreads();

  if (threadIdx.x < 16) {
    float s = 0.0f;
#pragma unroll
    for (int w = 0; w < 8; ++w) s += partial[w * 16 + threadIdx.x];
    const int n = col_base + threadIdx.x;
    const float z = (1.0f - tanhf((s + dope[n]) * 0.5f)) * 0.5f;
    M_out[2 * n + 0] = 1.0f - z;
    M_out[2 * n + 1] = z;
  }
}

// ---------------------------------------------------------------------------
extern "C" void kernel_launch(void* const* d_in, const int* in_sizes, int n_in,
                              void* d_out, int out_size, void* d_ws, size_t ws_size,
                              hipStream_t stream) {
  (void)in_sizes; (void)n_in; (void)out_size; (void)ws_size;
  const float* ps  = (const float*)d_in[0];   // [N,2]
  const float* Min = (const float*)d_in[1];   // [N,2]
  const float* Hxs = (const float*)d_in[2];   // [K,N]
  const int*   x   = (const int*)d_in[3];     // [K,1]
  const int*   H   = (const int*)d_in[4];     // [K,N]

  float* out      = (float*)d_out;
  float* M_out    = out;                      // [N,2] first
  float* Hxs_new  = out + (size_t)2 * NN;     // [K,N] second

  float* prior_clip = (float*)d_ws;           // N floats
  float* dope       = prior_clip + NN;        // N floats

  k1_prior<<<NN / 16, 256, 0, stream>>>(Hxs, ps, Min, prior_clip, dope);
  k2_rows<<<KK, 256, 0, stream>>>(Hxs, H, x, prior_clip, Hxs_new);
  k3_beliefs<<<NN / 16, 256, 0, stream>>>(Hxs_new, dope, M_out);
}